// AdjGAT_8091718386026
// MI455X (gfx1250) — compile-verified
//
#include <hip/hip_runtime.h>
#include <math.h>

// Problem constants from the reference
#define V_N    20000
#define D_K    256
#define K_NBR  16
#define O_DIM  128
#define H_HEADS 4
#define NEG_INF_F (-1e9f)

typedef float v2f __attribute__((ext_vector_type(2)));
typedef float v8f __attribute__((ext_vector_type(8)));

// ---------------------------------------------------------------------------
// Kernel 1: per-head GEMM  t[h] = x (V x 256) * W[h] (256 x 128)  via
// V_WMMA_F32_16X16X4_F32.  One wave computes one 16x16 tile of t[h];
// a 256-thread block (8 waves) covers the 8 column tiles (O=128) of one
// 16-row stripe for one head.  grid = (V/16, H).
//
// WMMA f32 16x16x4 VGPR layout (wave32):
//   A (16x4, MxK): lanes 0-15 hold rows M=0..15 with {K=0,K=1} in v0,v1;
//                  lanes 16-31 hold the same rows with {K=2,K=3}.
//   B (4x16, KxN): lanes 0-15 hold cols N=0..15 with {K=0,K=1} in v0,v1;
//                  lanes 16-31 hold the same cols with {K=2,K=3}.
//   C/D (16x16):   VGPR r: lanes 0-15 = D[M=r][N=lane],
//                          lanes 16-31 = D[M=r+8][N=lane-16].
// ---------------------------------------------------------------------------
__global__ __launch_bounds__(256) void gat_gemm_wmma_kernel(
    const float* __restrict__ x,   // [V, D]
    const float* __restrict__ W,   // [H, D, O]
    float* __restrict__ t)         // [H, V, O] workspace
{
    const int h        = blockIdx.y;
    const int row_base = blockIdx.x * 16;
    const int wave     = threadIdx.x >> 5;      // 0..7 -> column tile
    const int lane     = threadIdx.x & 31;
    const int half     = lane >> 4;             // 0: K pair {0,1}, 1: {2,3}
    const int l        = lane & 15;             // row (A) / col (B) index
    const int col_base = wave * 16;

    const float* xrow = x + (size_t)(row_base + l) * D_K;
    const float* Wh   = W + (size_t)h * D_K * O_DIM;

    v8f acc = {};
    #pragma unroll 4
    for (int k = 0; k < D_K; k += 4) {
        const int ka = k + 2 * half;            // this lane's first K element
        // A fragment: x[row_base+l][ka], x[row_base+l][ka+1]  (8B aligned)
        const float2 av = *reinterpret_cast<const float2*>(xrow + ka);
        v2f a_frag; a_frag.x = av.x; a_frag.y = av.y;
        // B fragment: W[h][ka][col_base+l], W[h][ka+1][col_base+l]
        v2f b_frag;
        b_frag.x = Wh[(size_t)ka * O_DIM + col_base + l];
        b_frag.y = Wh[(size_t)(ka + 1) * O_DIM + col_base + l];
        acc = __builtin_amdgcn_wmma_f32_16x16x4_f32(
            /*neg_a=*/false, a_frag, /*neg_b=*/false, b_frag,
            /*c_mod=*/(short)0, acc, /*reuse_a=*/false, /*reuse_b=*/false);
    }

    float* th = t + (size_t)h * V_N * O_DIM;
    #pragma unroll
    for (int r = 0; r < 8; ++r) {
        const int m = r + 8 * half;
        th[(size_t)(row_base + m) * O_DIM + col_base + l] = acc[r];
    }
}

// ---------------------------------------------------------------------------
// Kernel 2: attn[h,v] = dot(t[h,v,:], a[h,:])  -- one wave per (h,v) row.
// 32 lanes x float4 covers O=128 exactly; butterfly reduce within the wave.
// ---------------------------------------------------------------------------
__global__ __launch_bounds__(256) void gat_attn_kernel(
    const float* __restrict__ t,     // [H, V, O]
    const float* __restrict__ a,     // [H, O]
    float* __restrict__ attn)        // [H, V] workspace
{
    const int wave_global = (int)((blockIdx.x * blockDim.x + threadIdx.x) >> 5);
    if (wave_global >= H_HEADS * V_N) return;
    const int lane = threadIdx.x & 31;
    const int h = wave_global / V_N;

    const float4 tv = reinterpret_cast<const float4*>(
        t + (size_t)wave_global * O_DIM)[lane];
    const float4 av = reinterpret_cast<const float4*>(
        a + (size_t)h * O_DIM)[lane];

    float s = tv.x * av.x + tv.y * av.y + tv.z * av.z + tv.w * av.w;
    #pragma unroll
    for (int off = 16; off > 0; off >>= 1)
        s += __shfl_xor(s, off, 32);
    if (lane == 0) attn[wave_global] = s;
}

// ---------------------------------------------------------------------------
// Kernel 3: per node v -- gather neighbor logits, softmax over K=16,
// weighted sum of neighbor feature rows, mean over heads, + bias, ReLU.
// One block of 128 threads per node; thread tid owns output channel tid,
// so every neighbor-row read t[h][idx][tid] is a coalesced 512B line
// (L2-resident: t is 41 MB << 192 MB L2).
// ---------------------------------------------------------------------------
__global__ __launch_bounds__(128) void gat_aggregate_kernel(
    const float* __restrict__ t,        // [H, V, O]
    const float* __restrict__ attn,     // [H, V]
    const float* __restrict__ b,        // [H, O]
    const long long* __restrict__ adj,  // [V, K] int64
    const int* __restrict__ maskp,      // scalar mask_index (== V)
    float* __restrict__ out)            // [V, O]
{
    __shared__ int   s_idx[K_NBR];
    __shared__ float s_attn[K_NBR];

    const int v   = blockIdx.x;
    const int tid = threadIdx.x;

    if (tid < K_NBR) {
        const long long mi  = (long long)(*maskp);
        const long long idx = adj[(size_t)v * K_NBR + tid];
        s_idx[tid] = (idx == mi) ? -1 : (int)idx;   // -1 marks padding
    }
    __syncthreads();

    float acc_total = 0.f;
    for (int h = 0; h < H_HEADS; ++h) {
        if (tid < K_NBR) {
            const int idx = s_idx[tid];
            s_attn[tid] = (idx < 0) ? NEG_INF_F
                                    : attn[(size_t)h * V_N + idx];
        }
        __syncthreads();

        // softmax over K=16 (replicated per thread; 16 values from LDS)
        float mx = s_attn[0];
        #pragma unroll
        for (int k = 1; k < K_NBR; ++k) mx = fmaxf(mx, s_attn[k]);
        float w[K_NBR];
        float den = 0.f;
        #pragma unroll
        for (int k = 0; k < K_NBR; ++k) {
            w[k] = __expf(s_attn[k] - mx);
            den += w[k];
        }
        const float inv = 1.f / den;

        const float* th = t + (size_t)h * V_N * O_DIM;
        float acc = 0.f;
        #pragma unroll
        for (int k = 0; k < K_NBR; ++k) {
            const int idx = s_idx[k];           // uniform across block
            if (idx >= 0)
                acc += (w[k] * inv) * th[(size_t)idx * O_DIM + tid];
        }
        acc_total += acc + b[h * O_DIM + tid];
        __syncthreads();    // protect s_attn before next head rewrites it
    }

    out[(size_t)v * O_DIM + tid] =
        fmaxf(acc_total * (1.f / (float)H_HEADS), 0.f);
}

// ---------------------------------------------------------------------------
// Launch: t and attn live in d_ws (41.0 MB + 0.32 MB).
// Inputs (setup_inputs order): x, W, a, b, adj_lst(i64), mask_index(i32).
// ---------------------------------------------------------------------------
extern "C" void kernel_launch(void* const* d_in, const int* in_sizes, int n_in,
                              void* d_out, int out_size, void* d_ws, size_t ws_size,
                              hipStream_t stream) {
    const float*     x     = (const float*)d_in[0];
    const float*     W     = (const float*)d_in[1];
    const float*     a     = (const float*)d_in[2];
    const float*     b     = (const float*)d_in[3];
    const long long* adj   = (const long long*)d_in[4];
    const int*       maskp = (const int*)d_in[5];
    float*           out   = (float*)d_out;

    float* t    = (float*)d_ws;                              // H*V*O floats
    float* attn = t + (size_t)H_HEADS * V_N * O_DIM;         // H*V floats

    // 1) t = x * W[h]   (WMMA f32 16x16x4)
    dim3 g1(V_N / 16, H_HEADS);
    gat_gemm_wmma_kernel<<<g1, 256, 0, stream>>>(x, W, t);

    // 2) attn[h,v] = t[h,v,:] . a[h]
    const int n_rows = H_HEADS * V_N;                        // 80000 waves
    gat_attn_kernel<<<n_rows / 8, 256, 0, stream>>>(t, a, attn);

    // 3) softmax over neighbors + weighted aggregate + mean + ReLU
    gat_aggregate_kernel<<<V_N, 128, 0, stream>>>(t, attn, b, adj, maskp, out);
}